// RaggedSelectThreshold_59030030516971
// MI455X (gfx1250) — compile-verified
//
#include <hip/hip_runtime.h>
#include <stdint.h>

// SelectThreshold ragged compaction for MI455X (gfx1250).
// Pure data movement (~416 MB @ 23.3 TB/s ~= 18us floor). CDNA5 paths used:
//  - GLOBAL_LOAD_ASYNC_TO_LDS_B128 (ASYNCcnt) stages payload tiles in LDS while
//    the wave does ballot/scan bookkeeping; S_WAIT_ASYNCCNT + barrier, then
//    coalesced b128 drains.
//  - TH_STORE_NT on the 204.8 MB output stream (written once, never re-read)
//    so the read stream keeps the 192 MB L2.

#define TILE     128      // rows per tile (one block)
#define BLK      128      // threads per block (4 wave32)
#define NWAVES   (BLK/32)
#define SCAN_BLK 1024
#define MAX_SEG  64
#define MAX_F4   16       // F=64 -> 16 float4 per row (reference is fixed at F=64)

typedef float f4 __attribute__((ext_vector_type(4)));   // true vector type (NT-store OK)

// ---------------- kernel: async-stage tile in LDS, scatter compacted rows ----
// (defined first so the disasm snippet shows the async-LDS lowering)
__global__ void scatter_kernel(const float* __restrict__ xs,
                               const float* __restrict__ pl,
                               const int* __restrict__ tile_offsets,
                               int N, int F4, int F4shift,
                               f4* __restrict__ out_feat,
                               int* __restrict__ out_scatter) {
    __shared__ f4  tileLds[TILE * MAX_F4];   // 32 KB payload stage
    __shared__ int rowlist[TILE];
    __shared__ int wsum[NWAVES];

    int t     = threadIdx.x;
    int tile  = blockIdx.x;
    int base  = tile * TILE;
    int valid = min(TILE, N - base);

    // LDS aperture: flat addr bits [31:0] == group-relative LDS byte offset.
    unsigned ldsbase = (unsigned)(uintptr_t)&tileLds[0];
    uint64_t gbase   = (uint64_t)(uintptr_t)pl + ((uint64_t)base * (uint64_t)F4) * 16ull;

    // Stage the whole tile into LDS asynchronously (per-lane EXEC gates tail).
    int total4 = valid * F4;
    for (int e = t; e < total4; e += BLK) {
        unsigned loff = ldsbase + (unsigned)e * 16u;
        uint64_t ga   = gbase + (uint64_t)e * 16ull;
        asm volatile("global_load_async_to_lds_b128 %0, %1, off"
                     :: "v"(loff), "v"(ga) : "memory");
    }

    // Overlap: stable intra-tile positions via ballot prefix + wave sums.
    int row = base + t;
    int m   = (t < valid) ? ((xs[row] > 0.5f) ? 1 : 0) : 0;

    unsigned long long bal = __ballot(m);
    int lane = t & 31, w = t >> 5;
    int wpre = __popcll(bal & ((1ull << lane) - 1ull));
    if (lane == 0) wsum[w] = __popcll(bal);
    __syncthreads();

    int woff = 0, tot = 0;
    for (int i = 0; i < NWAVES; ++i) { if (i < w) woff += wsum[i]; tot += wsum[i]; }
    int lpos = woff + wpre;
    int toff = tile_offsets[tile];

    if (m) {
        rowlist[lpos] = t;                 // tile-local source row
        out_scatter[toff + lpos] = row;    // original row id (int32 bits)
    }

    // Wait for this wave's async transfers, then block barrier.
    asm volatile("s_wait_asynccnt 0x0" ::: "memory");
    __syncthreads();

    // Drain the `tot` selected rows LDS -> global; coalesced NT b128 stores.
    int cnt4 = tot * F4;
    for (int k = t; k < cnt4; k += BLK) {
        int j = k >> F4shift;              // which selected row
        int c = k - (j << F4shift);        // float4 column
        int r = rowlist[j];
        f4 v  = tileLds[(r << F4shift) + c];
        __builtin_nontemporal_store(v, &out_feat[((size_t)(toff + j) << F4shift) + c]);
    }
}

// ---------------- kernel: zero segment counters in workspace -----------------
__global__ void zero_ws_kernel(int* seg_counts, int B) {
    int t = threadIdx.x;
    if (t < B) seg_counts[t] = 0;
}

// ---------------- kernel: per-tile kept counts + per-segment counts ----------
__global__ void count_kernel(const float* __restrict__ xs,
                             const int* __restrict__ rs,
                             int N, int B,
                             int* __restrict__ tile_counts,
                             int* __restrict__ seg_counts) {
    __shared__ int srs[MAX_SEG + 1];
    __shared__ int scnt[MAX_SEG];
    __shared__ int wsum[NWAVES];

    int t    = threadIdx.x;
    int tile = blockIdx.x;
    int row  = tile * TILE + t;

    if (t <= B) srs[t]  = rs[t];
    if (t <  B) scnt[t] = 0;
    __syncthreads();

    int m = 0;
    if (row < N) m = (xs[row] > 0.5f) ? 1 : 0;

    unsigned long long bal = __ballot(m);   // wave32: low 32 bits
    int lane = t & 31, w = t >> 5;
    if (lane == 0) wsum[w] = __popcll(bal);

    if (m) {
        // searchsorted(rs, row, 'right') - 1  (B is small: linear scan)
        int b = 0;
        for (int i = 1; i < B; ++i) if (row >= srs[i]) b = i;
        atomicAdd(&scnt[b], 1);
    }
    __syncthreads();

    if (t == 0) {
        int s = 0;
        for (int i = 0; i < NWAVES; ++i) s += wsum[i];
        tile_counts[tile] = s;
    }
    if (t < B) {
        int v = scnt[t];
        if (v) atomicAdd(&seg_counts[t], v);   // integer sum: order-independent
    }
}

// ---------------- kernel: scan tile counts, emit new_rs ----------------------
__global__ void scan_kernel(const int* __restrict__ tile_counts, int ntiles,
                            int* __restrict__ tile_offsets,
                            int* __restrict__ n_sel,
                            const int* __restrict__ seg_counts, int B,
                            int* __restrict__ out_rs) {
    __shared__ int ssum[SCAN_BLK];
    int t     = threadIdx.x;
    int chunk = (ntiles + SCAN_BLK - 1) / SCAN_BLK;
    int beg   = t * chunk;
    int end   = min(beg + chunk, ntiles);

    int s = 0;
    for (int i = beg; i < end; ++i) s += tile_counts[i];
    ssum[t] = s;
    __syncthreads();

    for (int off = 1; off < SCAN_BLK; off <<= 1) {   // inclusive Hillis-Steele
        int v = (t >= off) ? ssum[t - off] : 0;
        __syncthreads();
        ssum[t] += v;
        __syncthreads();
    }

    int run = ssum[t] - s;                            // exclusive prefix
    for (int i = beg; i < end; ++i) { tile_offsets[i] = run; run += tile_counts[i]; }

    if (t == 0) {
        n_sel[0] = ssum[SCAN_BLK - 1];
        int acc = 0;
        out_rs[0] = 0;
        for (int b = 0; b < B; ++b) { acc += seg_counts[b]; out_rs[b + 1] = acc; }
    }
}

// ---------------- kernel: zero-pad tail rows, pad scatter idx with N ---------
__global__ void tail_kernel(const int* __restrict__ n_sel_p, int N, int F4shift,
                            f4* __restrict__ out_feat,
                            int* __restrict__ out_scatter) {
    long idx   = (long)blockIdx.x * blockDim.x + threadIdx.x;
    long total = (long)N << F4shift;
    if (idx >= total) return;
    int  nsel = *n_sel_p;
    long row  = idx >> F4shift;
    if (row >= nsel) {
        f4 z = (f4){0.f, 0.f, 0.f, 0.f};
        __builtin_nontemporal_store(z, &out_feat[idx]);
    }
    if (idx < N && idx >= nsel) out_scatter[idx] = N;
}

// ---------------- host launcher ----------------------------------------------
extern "C" void kernel_launch(void* const* d_in, const int* in_sizes, int n_in,
                              void* d_out, int out_size, void* d_ws, size_t ws_size,
                              hipStream_t stream) {
    const float* xs = (const float*)d_in[0];
    const float* pl = (const float*)d_in[1];
    const int*   rs = (const int*)d_in[2];

    int N = in_sizes[0];            // 800000
    int F = in_sizes[1] / N;        // 64
    int B = in_sizes[2] - 1;        // 8
    int F4 = F / 4;                 // 16
    int F4shift = 0; while ((1 << F4shift) < F4) ++F4shift;

    int ntiles = (N + TILE - 1) / TILE;

    // Output layout (flat, return order): newfeat [N*F] f32, new_rs [B+1] i32,
    // scatter_idxs [N] i32 (int values stored bit-exact into the buffer).
    float* out_feat    = (float*)d_out;
    int*   out_rs      = (int*)(out_feat + (size_t)N * F);
    int*   out_scatter = out_rs + (B + 1);

    int* ws           = (int*)d_ws;
    int* tile_counts  = ws;                    // ntiles
    int* tile_offsets = ws + ntiles;           // ntiles
    int* seg_counts   = ws + 2 * ntiles;       // B
    int* n_sel        = ws + 2 * ntiles + B;   // 1

    zero_ws_kernel<<<1, 256, 0, stream>>>(seg_counts, B);
    count_kernel<<<ntiles, BLK, 0, stream>>>(xs, rs, N, B, tile_counts, seg_counts);
    scan_kernel<<<1, SCAN_BLK, 0, stream>>>(tile_counts, ntiles, tile_offsets,
                                            n_sel, seg_counts, B, out_rs);
    scatter_kernel<<<ntiles, BLK, 0, stream>>>(xs, pl, tile_offsets, N, F4, F4shift,
                                               (f4*)out_feat, out_scatter);

    long total4 = (long)N << F4shift;
    int  tb = 256;
    long tg = (total4 + tb - 1) / tb;
    tail_kernel<<<(int)tg, tb, 0, stream>>>(n_sel, N, F4shift,
                                            (f4*)out_feat, out_scatter);
}